// TripletLoss_14998025797938
// MI455X (gfx1250) — compile-verified
//
#include <hip/hip_runtime.h>
#include <math.h>
#include <stdint.h>

// ---------------------------------------------------------------------------
// TripletLoss with batch-hard mining + DC cross-attention alignment (gfx1250)
//   B=1024, C=1, W=H=128, NUM_IDS=64, MARGIN=0.3
// Matrix work: V_WMMA_F32_16X16X4_F32 (f32 == reference precision).
// Data movement: GLOBAL_LOAD_ASYNC_TO_LDS_B128 (ASYNCcnt) with double
// buffering in the gram GEMM.
// ---------------------------------------------------------------------------

typedef __attribute__((ext_vector_type(2))) float v2f;
typedef __attribute__((ext_vector_type(8))) float v8f;

#define N_SAMPLES 1024
#define DIM 16384      // 128*128 flattened feature
#define WH 128
#define MARGIN 0.3f

static __device__ __forceinline__ v8f wmma_f32(v2f a, v2f b, v8f c) {
  // 8 args: (neg_a, A, neg_b, B, c_mod, C, reuse_a, reuse_b)
  return __builtin_amdgcn_wmma_f32_16x16x4_f32(false, a, false, b, (short)0, c,
                                               false, false);
}

// Async 16-byte copy global -> LDS (per lane). lds_addr = raw LDS byte addr
// (low 32 bits of the generic pointer, per the flat-aperture mapping).
static __device__ __forceinline__ void async_copy16(unsigned lds_addr,
                                                    const float* gptr) {
  asm volatile("global_load_async_to_lds_b128 %0, %1, off"
               :: "v"(lds_addr), "v"(gptr)
               : "memory");
}

static __device__ __forceinline__ void wait_async0() {
  asm volatile("s_wait_asynccnt 0x0" ::: "memory");
}

static __device__ __forceinline__ unsigned lds_addr_of(const void* p) {
  return (unsigned)(uintptr_t)p;
}

// ---------------------------------------------------------------------------
// Kernel 1: Gram matrix G = X @ X^T  (1024 x 1024, K = 16384)
// 128x64 output tile per 256-thread block; 8 waves, each wave owns a 2x2
// block of 16x16 subtiles (4 WMMAs per 4 LDS fragment loads).
// K-slices (128x32 + 64x32) staged via async global->LDS copies, double
// buffered: issue slice k+1 while WMMAs consume slice k.
// ---------------------------------------------------------------------------
#define GM_M 128
#define GM_N 64
#define KT 32
#define APAD 33  // stride padding -> conflict-free fragment loads

__global__ __launch_bounds__(256) void tl_gram(const float* __restrict__ X,
                                               float* __restrict__ G) {
  __shared__ float As[2][GM_M * APAD];
  __shared__ float Bs[2][GM_N * APAD];
  const int tid = threadIdx.x;
  const int wave = tid >> 5, lane = tid & 31;
  const int laneM = lane & 15, kb = (lane >> 4) << 1;  // K-pair select
  const int rowBase = blockIdx.y * GM_M, colBase = blockIdx.x * GM_N;
  const int wm = wave & 3;   // row group: 32 rows
  const int wn = wave >> 2;  // col group: 32 cols
  v8f acc00 = {}, acc01 = {}, acc10 = {}, acc11 = {};

  // Async staging: A slice = 128 rows x 8 chunks(16B) = 1024 chunks (4/thread)
  //                B slice =  64 rows x 8 chunks      =  512 chunks (2/thread)
  const int srow = tid >> 3;          // 0..31 base row (chunk id / 8)
  const int sch  = (tid & 7) << 2;    // float offset of 16B chunk in row
  // Re-derive per-chunk row from chunk id c = tid + i*256: row = c>>3.

#define ISSUE_SLICE(buf, k0)                                                   \
  do {                                                                         \
    _Pragma("unroll")                                                          \
    for (int i_ = 0; i_ < 4; ++i_) {                                           \
      int c_ = tid + i_ * 256;                                                 \
      int r_ = c_ >> 3, ch_ = (c_ & 7) << 2;                                   \
      async_copy16(lds_addr_of(&As[buf][r_ * APAD + ch_]),                     \
                   X + (size_t)(rowBase + r_) * DIM + (k0) + ch_);             \
    }                                                                          \
    _Pragma("unroll")                                                          \
    for (int i_ = 0; i_ < 2; ++i_) {                                           \
      int c_ = tid + i_ * 256;                                                 \
      int r_ = c_ >> 3, ch_ = (c_ & 7) << 2;                                   \
      async_copy16(lds_addr_of(&Bs[buf][r_ * APAD + ch_]),                     \
                   X + (size_t)(colBase + r_) * DIM + (k0) + ch_);             \
    }                                                                          \
  } while (0)

  (void)srow; (void)sch;
  ISSUE_SLICE(0, 0);

  for (int k0 = 0; k0 < DIM; k0 += KT) {
    const int buf = (k0 >> 5) & 1;
    wait_async0();       // this wave's async copies for `buf` have landed
    __syncthreads();     // everyone's copies visible; prior compute done
    if (k0 + KT < DIM) ISSUE_SLICE(buf ^ 1, k0 + KT);

    const float* a0 = &As[buf][(wm * 32 + laneM) * APAD];
    const float* a1 = &As[buf][(wm * 32 + 16 + laneM) * APAD];
    const float* b0 = &Bs[buf][(wn * 32 + laneM) * APAD];
    const float* b1 = &Bs[buf][(wn * 32 + 16 + laneM) * APAD];
#pragma unroll
    for (int kk = 0; kk < KT; kk += 4) {
      v2f fa0, fa1, fb0, fb1;
      fa0.x = a0[kk + kb]; fa0.y = a0[kk + kb + 1];
      fa1.x = a1[kk + kb]; fa1.y = a1[kk + kb + 1];
      fb0.x = b0[kk + kb]; fb0.y = b0[kk + kb + 1];
      fb1.x = b1[kk + kb]; fb1.y = b1[kk + kb + 1];
      acc00 = wmma_f32(fa0, fb0, acc00);
      acc01 = wmma_f32(fa0, fb1, acc01);
      acc10 = wmma_f32(fa1, fb0, acc10);
      acc11 = wmma_f32(fa1, fb1, acc11);
    }
  }
#undef ISSUE_SLICE

  // C/D layout: N = lane&15; VGPR i holds M = i + 8*(lane>=16)
  const int m0 = rowBase + wm * 32 + ((lane >> 4) << 3);
  const int c0 = colBase + wn * 32 + laneM;
#pragma unroll
  for (int i = 0; i < 8; ++i) {
    G[(size_t)(m0 + i) * N_SAMPLES + c0]       = acc00[i];
    G[(size_t)(m0 + i) * N_SAMPLES + c0 + 16]  = acc01[i];
    G[(size_t)(m0 + 16 + i) * N_SAMPLES + c0]      = acc10[i];
    G[(size_t)(m0 + 16 + i) * N_SAMPLES + c0 + 16] = acc11[i];
  }
}

// ---------------------------------------------------------------------------
// Kernel 2: diag[i] = G[i][i] (squared norms)
// ---------------------------------------------------------------------------
__global__ __launch_bounds__(256) void tl_diag(const float* __restrict__ G,
                                               float* __restrict__ diag) {
  int i = blockIdx.x * 256 + threadIdx.x;
  if (i < N_SAMPLES) diag[i] = G[(size_t)i * N_SAMPLES + i];
}

// ---------------------------------------------------------------------------
// Kernel 3: batch-hard mining per row; first-index tie-break matches
// jnp.argmax/argmin semantics.
// ---------------------------------------------------------------------------
__global__ __launch_bounds__(256) void tl_mine(const float* __restrict__ G,
                                               const float* __restrict__ diag,
                                               const int* __restrict__ tgt,
                                               int* __restrict__ hp,
                                               int* __restrict__ hn) {
  __shared__ float pv_s[256]; __shared__ int pi_s[256];
  __shared__ float nv_s[256]; __shared__ int ni_s[256];
  const int i = blockIdx.x;
  const int tid = threadIdx.x;
  const int ti = tgt[i];
  const float di = diag[i];
  float pv = -1e30f; int pi = N_SAMPLES;
  float nv =  1e30f; int ni = N_SAMPLES;
  for (int j = tid; j < N_SAMPLES; j += 256) {
    float d2 = di + diag[j] - 2.f * G[(size_t)i * N_SAMPLES + j];
    float d = sqrtf(fmaxf(d2, 1e-12f));
    if (tgt[j] == ti) { if (d > pv) { pv = d; pi = j; } }
    else              { if (d < nv) { nv = d; ni = j; } }
  }
  pv_s[tid] = pv; pi_s[tid] = pi; nv_s[tid] = nv; ni_s[tid] = ni;
  __syncthreads();
  for (int s = 128; s > 0; s >>= 1) {
    if (tid < s) {
      if (pv_s[tid + s] > pv_s[tid] ||
          (pv_s[tid + s] == pv_s[tid] && pi_s[tid + s] < pi_s[tid])) {
        pv_s[tid] = pv_s[tid + s]; pi_s[tid] = pi_s[tid + s];
      }
      if (nv_s[tid + s] < nv_s[tid] ||
          (nv_s[tid + s] == nv_s[tid] && ni_s[tid + s] < ni_s[tid])) {
        nv_s[tid] = nv_s[tid + s]; ni_s[tid] = ni_s[tid + s];
      }
    }
    __syncthreads();
  }
  if (tid == 0) { hp[i] = pi_s[0]; hn[i] = ni_s[0]; }
}

// ---------------------------------------------------------------------------
// Kernel 4: DC alignment + fused squared distance, one block per (b, branch).
//  S = (A @ O^T)/sqrt(128); P = row_softmax(S) in place; modi = P @ O;
//  dist2 = sum((A - modi)^2). ~195 KB of the 320 KB WGP LDS; anchor/other
//  tiles staged with async global->LDS copies.
// ---------------------------------------------------------------------------
#define LDP 130
__global__ __launch_bounds__(256) void tl_dc(const float* __restrict__ X,
                                             const int* __restrict__ hp,
                                             const int* __restrict__ hn,
                                             float* __restrict__ dist2) {
  __shared__ float Anc[WH * LDP];
  __shared__ float Oth[WH * LDP];
  __shared__ float Sm [WH * LDP];
  __shared__ float red[8];
  const int b = blockIdx.x;
  const int br = blockIdx.y;  // 0 = hard positive, 1 = hard negative
  const int o = (br == 0) ? hp[b] : hn[b];
  const int tid = threadIdx.x;
  const int wave = tid >> 5, lane = tid & 31;
  const int laneM = lane & 15, kb = (lane >> 4) << 1;

  // stage anchor + other (64 KB each) via async copies: 128x8 16B chunks each
  {
    const float* Ag = X + (size_t)b * DIM;
    const float* Og = X + (size_t)o * DIM;
    const unsigned ancB = lds_addr_of(&Anc[0]);
    const unsigned othB = lds_addr_of(&Oth[0]);
#pragma unroll
    for (int i = 0; i < 4; ++i) {
      int c = tid + i * 256;           // chunk id 0..1023
      int r = c >> 3, ch = (c & 7) << 2;
      unsigned off = (unsigned)((r * LDP + ch) * 4);
      async_copy16(ancB + off, Ag + r * WH + ch);
      async_copy16(othB + off, Og + r * WH + ch);
    }
  }
  wait_async0();
  __syncthreads();

  const float scale = 0.08838834764831845f;  // 1/sqrt(128)

  // Phase 1: S = A @ O^T  (B fragment = rows of O, since B[k][n] = O[n][k])
  for (int s = wave; s < 64; s += 8) {
    const int sm = s >> 3, sn = s & 7;
    v8f acc = {};
    const float* ar = Anc + (sm * 16 + laneM) * LDP;
    const float* orow = Oth + (sn * 16 + laneM) * LDP;
#pragma unroll 8
    for (int kk = 0; kk < WH; kk += 4) {
      v2f a, bb;
      a.x = ar[kk + kb];    a.y = ar[kk + kb + 1];
      bb.x = orow[kk + kb]; bb.y = orow[kk + kb + 1];
      acc = wmma_f32(a, bb, acc);
    }
    const int mrow = sm * 16 + ((lane >> 4) << 3);
    const int col = sn * 16 + laneM;
#pragma unroll
    for (int i = 0; i < 8; ++i) Sm[(mrow + i) * LDP + col] = acc[i] * scale;
  }
  __syncthreads();

  // Phase 2: in-place row softmax of S
  if (tid < WH) {
    float* row = Sm + tid * LDP;
    float m = row[0];
    for (int j = 1; j < WH; ++j) m = fmaxf(m, row[j]);
    float ssum = 0.f;
    for (int j = 0; j < WH; ++j) { float e = __expf(row[j] - m); row[j] = e; ssum += e; }
    float inv = 1.f / ssum;
    for (int j = 0; j < WH; ++j) row[j] *= inv;
  }
  __syncthreads();

  // Phase 3: modi = P @ O, fused with sum((A - modi)^2)
  float ssq = 0.f;
  for (int s = wave; s < 64; s += 8) {
    const int sm = s >> 3, sn = s & 7;
    v8f acc = {};
    const float* ar = Sm + (sm * 16 + laneM) * LDP;
    const int col = sn * 16 + laneM;
#pragma unroll 8
    for (int kk = 0; kk < WH; kk += 4) {
      v2f a, bb;
      a.x = ar[kk + kb]; a.y = ar[kk + kb + 1];
      bb.x = Oth[(kk + kb) * LDP + col];
      bb.y = Oth[(kk + kb + 1) * LDP + col];
      acc = wmma_f32(a, bb, acc);
    }
    const int mrow = sm * 16 + ((lane >> 4) << 3);
#pragma unroll
    for (int i = 0; i < 8; ++i) {
      float dd = Anc[(mrow + i) * LDP + col] - acc[i];
      ssq += dd * dd;
    }
  }
  // block reduce (wave32 shuffles + LDS)
  for (int off = 16; off > 0; off >>= 1) ssq += __shfl_down(ssq, off, 32);
  if (lane == 0) red[wave] = ssq;
  __syncthreads();
  if (tid == 0) {
    float t = 0.f;
    for (int w = 0; w < 8; ++w) t += red[w];
    dist2[br * N_SAMPLES + b] = t;
  }
}

// ---------------------------------------------------------------------------
// Kernel 5: loss = mean(relu(sqrt(d2_ap) - sqrt(d2_an) + margin))
// ---------------------------------------------------------------------------
__global__ __launch_bounds__(256) void tl_loss(const float* __restrict__ dist2,
                                               float* __restrict__ out) {
  __shared__ float red[8];
  const int tid = threadIdx.x;
  float acc = 0.f;
  for (int i = tid; i < N_SAMPLES; i += 256) {
    float ap = sqrtf(dist2[i]);
    float an = sqrtf(dist2[N_SAMPLES + i]);
    acc += fmaxf(ap - an + MARGIN, 0.f);
  }
  for (int off = 16; off > 0; off >>= 1) acc += __shfl_down(acc, off, 32);
  if ((tid & 31) == 0) red[tid >> 5] = acc;
  __syncthreads();
  if (tid == 0) {
    float t = 0.f;
    for (int w = 0; w < 8; ++w) t += red[w];
    out[0] = t / (float)N_SAMPLES;
  }
}

// ---------------------------------------------------------------------------
extern "C" void kernel_launch(void* const* d_in, const int* in_sizes, int n_in,
                              void* d_out, int out_size, void* d_ws, size_t ws_size,
                              hipStream_t stream) {
  const float* X = (const float*)d_in[0];   // (1024, 1, 128, 128) f32
  const int* tgt = (const int*)d_in[1];     // (1024,) int32
  float* out = (float*)d_out;               // scalar f32 loss

  // workspace layout
  char* ws = (char*)d_ws;
  float* G    = (float*)ws;                                       // 4 MB gram
  float* diag = (float*)(ws + (size_t)N_SAMPLES * N_SAMPLES * 4); // 4 KB
  int*   hp   = (int*)(diag + N_SAMPLES);                         // 4 KB
  int*   hn   = hp + N_SAMPLES;                                   // 4 KB
  float* dist2 = (float*)(hn + N_SAMPLES);                        // 8 KB

  tl_gram<<<dim3(N_SAMPLES / GM_N, N_SAMPLES / GM_M), 256, 0, stream>>>(X, G);
  tl_diag<<<N_SAMPLES / 256, 256, 0, stream>>>(G, diag);
  tl_mine<<<N_SAMPLES, 256, 0, stream>>>(G, diag, tgt, hp, hn);
  tl_dc<<<dim3(N_SAMPLES, 2), 256, 0, stream>>>(X, hp, hn, dist2);
  tl_loss<<<1, 256, 0, stream>>>(dist2, out);
}